// AdaptiveLIFLayer_87239375717047
// MI455X (gfx1250) — compile-verified
//
#include <hip/hip_runtime.h>

// LIF scan: out[b,t,n] = spike of v-recurrence, v' = 0.5*(v + x); spike => hard reset.
// Purely bandwidth-bound (420 MB traffic, ~18 us floor at 23.3 TB/s).
// CDNA5 path: GLOBAL_LOAD_ASYNC_TO_LDS_B128, 8-deep pipeline via ASYNCcnt,
// ds_load_b128 stage reads, nontemporal B128 stores.

typedef __attribute__((ext_vector_type(4))) float fvec4;

#define T_STEPS 200
#define N_NEUR  4096
#define B_BATCH 64
#define DEPTH   8                      // async pipeline depth (power of 2)
#define TPB     256                    // 8 waves per block (wave32)
#define CHUNK   (TPB * 4)              // 1024 neurons per block
#define BLOCKS  (B_BATCH * (N_NEUR / CHUNK))   // 64 * 4 = 256 blocks

__global__ __launch_bounds__(TPB) void lif_scan_kernel(
    const float* __restrict__ x, float* __restrict__ out) {
  // Single shared array -> allocated at LDS offset 0; the inline asm below
  // computes raw LDS byte offsets against that base.
  __shared__ __attribute__((aligned(16))) float smem[DEPTH * CHUNK];  // 32 KB

  const int tid   = threadIdx.x;
  const int bid   = blockIdx.x;
  const int b     = bid >> 2;      // batch index
  const int chunk = bid & 3;       // which quarter of the N dimension
  const int n0    = chunk * CHUNK + tid * 4;

  const float* __restrict__ xb = x   + (size_t)b * T_STEPS * N_NEUR;
  float* __restrict__       ob = out + (size_t)b * T_STEPS * N_NEUR;

  // Per-lane global byte offset within a time-step row (GVS mode VADDR).
  const unsigned goff = (unsigned)(n0 * 4);

  // Issue one async global->LDS 128-bit load (16 B per lane) for time step t
  // into pipeline stage `stage`. SADDR = uniform row pointer (SGPR pair),
  // VADDR = per-lane byte offset, VDST = per-lane LDS byte address.
  // NOTE: &smem[0] is passed as an (unreferenced) operand so the shared
  // array's address escapes into the asm -- without this, LLVM proves the
  // asm cannot write smem and deletes the ds_load reads entirely.
  auto issue = [&](int t, int stage) {
    const unsigned lds_off = (unsigned)(stage * (CHUNK * 4) + tid * 16);
    const float* src = xb + (size_t)t * N_NEUR;  // uniform -> SGPR pair
    asm volatile("global_load_async_to_lds_b128 %0, %1, %2"
                 :
                 : "v"(lds_off), "v"(goff), "s"(src), "v"(&smem[0])
                 : "memory");
  };

  fvec4 v = {0.0f, 0.0f, 0.0f, 0.0f};

  // One LIF step: update v, produce spikes, stream them out.
  auto step = [&](int t, int stage) {
    const fvec4 x4 =
        *reinterpret_cast<const fvec4*>(&smem[stage * CHUNK + tid * 4]);
    // v += (x - v)/tau with tau=2, v_reset=0  ->  v = 0.5*(v + x)
    v = 0.5f * (v + x4);
    fvec4 s;
    s.x = (v.x >= 1.0f) ? 1.0f : 0.0f;
    s.y = (v.y >= 1.0f) ? 1.0f : 0.0f;
    s.z = (v.z >= 1.0f) ? 1.0f : 0.0f;
    s.w = (v.w >= 1.0f) ? 1.0f : 0.0f;
    // hard reset
    v.x = (v.x >= 1.0f) ? 0.0f : v.x;
    v.y = (v.y >= 1.0f) ? 0.0f : v.y;
    v.z = (v.z >= 1.0f) ? 0.0f : v.z;
    v.w = (v.w >= 1.0f) ? 0.0f : v.w;
    // Streamed write-once output: nontemporal 128-bit store.
    __builtin_nontemporal_store(
        s, reinterpret_cast<fvec4*>(ob + (size_t)t * N_NEUR + n0));
  };

  // Prologue: fill the pipeline (stages 0..7 <- t = 0..7).
#pragma unroll
  for (int t = 0; t < DEPTH; ++t) issue(t, t);

  // Steady state: exactly DEPTH loads outstanding at the top of every
  // iteration; async loads complete in order, so <= DEPTH-1 outstanding
  // means the stage we are about to read has landed in LDS.
  // T_STEPS - DEPTH = 192 = 24 * 8: unroll by DEPTH keeps stages constant.
#pragma unroll 8
  for (int t = 0; t < T_STEPS - DEPTH; ++t) {
    asm volatile("s_wait_asynccnt 7" ::: "memory");
    const int stage = t & (DEPTH - 1);
    // Read the stage, then make sure it drained into VGPRs before refilling.
    const fvec4* slot =
        reinterpret_cast<const fvec4*>(&smem[stage * CHUNK + tid * 4]);
    (void)slot;
    // step() performs the ds_load; order the refill after it:
    step(t, stage);
    asm volatile("s_wait_dscnt 0" ::: "memory");
    issue(t + DEPTH, stage);
  }

  // Drain epilogue: wait for everything once, then consume the last
  // DEPTH stages with no refills (a wait of DEPTH-1 here would NOT
  // guarantee completion since up to DEPTH loads remain in flight).
  asm volatile("s_wait_asynccnt 0" ::: "memory");
#pragma unroll
  for (int t = T_STEPS - DEPTH; t < T_STEPS; ++t) {
    step(t, t & (DEPTH - 1));
  }
}

extern "C" void kernel_launch(void* const* d_in, const int* in_sizes, int n_in,
                              void* d_out, int out_size, void* d_ws, size_t ws_size,
                              hipStream_t stream) {
  const float* x = (const float*)d_in[0];
  // d_in[1] (threshold) is part of the module but unused in the math.
  float* out = (float*)d_out;
  lif_scan_kernel<<<dim3(BLOCKS), dim3(TPB), 0, stream>>>(x, out);
}